// CBOWNeg_22368189677637
// MI455X (gfx1250) — compile-verified
//
#include <hip/hip_runtime.h>

// CBOW negative-sampling loss on gfx1250.
// One wave32 block handles 16 examples; scores computed with chained
// V_WMMA_F32_16X16X4_F32 (D = A*B + C), diagonal extracted per lane.

typedef float v2f __attribute__((ext_vector_type(2)));
typedef float v4f __attribute__((ext_vector_type(4)));
typedef float v8f __attribute__((ext_vector_type(8)));

#define DIM        128
#define CTXN       8
#define KNEG       10
#define GRP        16          // examples per wave
#define LDS_STRIDE 132         // floats per v-row: 528B -> 16B aligned, bank-conflict-free frags

__device__ __forceinline__ float log_sigmoid_f(float x) {
    // logsig(x) = min(x,0) - log1p(exp(-|x|))  (numerically stable)
    return fminf(x, 0.0f) - log1pf(__expf(-fabsf(x)));
}

__global__ __launch_bounds__(32)
void cbow_neg_wmma_kernel(const float* __restrict__ in_embed,
                          const float* __restrict__ out_embed,
                          const int*   __restrict__ context,
                          const int*   __restrict__ target,
                          const int*   __restrict__ negatives,
                          float*       __restrict__ out)
{
    __shared__ float v_lds[GRP * LDS_STRIDE];   // 8448 B

    const int lane = threadIdx.x;               // 0..31 (wave32)
    const int b0   = blockIdx.x * GRP;

    // ---------- Phase 1: v_i = mean_c in_embed[context[b0+i][c]]  -> LDS ----------
    // Each lane owns a float4 slice of the 128-dim row: dims [lane*4, lane*4+4).
    for (int i = 0; i < GRP; ++i) {
        const int* cp = context + (size_t)(b0 + i) * CTXN;   // uniform -> s_load
        v4f acc = {0.0f, 0.0f, 0.0f, 0.0f};
        #pragma unroll
        for (int c = 0; c < CTXN; ++c) {
            const int row = cp[c];                           // uniform scalar
            const v4f* src = (const v4f*)(in_embed + (size_t)row * DIM);
            v4f x = src[lane];                               // global_load_b128, coalesced
            acc += x;
        }
        acc *= 0.125f;                                       // /CTXN
        *((v4f*)(v_lds + i * LDS_STRIDE) + lane) = acc;      // ds_store_b128
    }
    __syncthreads();   // single-wave WG: barrier degenerates, but orders LDS

    // ---------- Phase 2: hoist all A-matrix fragments (V, 16x128) into VGPRs ----------
    // A-frag layout (32b A 16x4): lanes 0-15 -> M=lane, K={k0,k0+1};
    //                             lanes 16-31 -> M=lane-16, K={k0+2,k0+3}.
    const int col = lane & 15;                 // matrix row/col handled by this lane
    const int ko  = (lane < 16) ? 0 : 2;
    v2f afrag[32];
    #pragma unroll
    for (int kk = 0; kk < 32; ++kk) {
        afrag[kk] = *(const v2f*)(v_lds + col * LDS_STRIDE + kk * 4 + ko); // ds_load_b64
    }

    // Diagonal ownership: C/D 16x16 f32 layout:
    //   vgpr r, lanes 0-15  -> (M=r,   N=lane)
    //   vgpr r, lanes 16-31 -> (M=8+r, N=lane-16)
    // => diag i in 0..7  lives at lane i,    c[i]
    //    diag i in 8..15 lives at lane i+16, c[i-8]
    const bool valid = (lane < 8) | (lane >= 24);
    const int  r     = (lane < 8) ? lane : (lane - 24);   // c[] index for valid lanes

    float acc_loss = 0.0f;

    for (int s = 0; s < 1 + KNEG; ++s) {
        // per-lane output-embedding row for this lane's column (example b0+col)
        const int urow = (s == 0) ? target[b0 + col]
                                  : negatives[(size_t)(b0 + col) * KNEG + (s - 1)];
        const float* ub = out_embed + (size_t)urow * DIM + ko;

        v8f c = {};   // C = 0
        #pragma unroll
        for (int kk = 0; kk < 32; ++kk) {
            v2f bfrag = *(const v2f*)(ub + kk * 4);          // global_load_b64 gather
            // D = A x B + C   (8 args: neg_a, A, neg_b, B, c_mod, C, reuse_a, reuse_b)
            c = __builtin_amdgcn_wmma_f32_16x16x4_f32(false, afrag[kk],
                                                      false, bfrag,
                                                      (short)0, c,
                                                      false, false);
        }

        // branch-free diagonal extraction (cndmask chain; EXEC stays all-ones)
        float d = c[0];
        d = (r == 1) ? c[1] : d;
        d = (r == 2) ? c[2] : d;
        d = (r == 3) ? c[3] : d;
        d = (r == 4) ? c[4] : d;
        d = (r == 5) ? c[5] : d;
        d = (r == 6) ? c[6] : d;
        d = (r == 7) ? c[7] : d;

        const float x = (s == 0) ? d : -d;   // pos: logsig(score); neg: logsig(-score)
        acc_loss += log_sigmoid_f(x);
    }

    if (valid) {
        const int i = (lane < 8) ? lane : (lane - 16);   // example within group
        out[b0 + i] = -acc_loss;
    }
}

extern "C" void kernel_launch(void* const* d_in, const int* in_sizes, int n_in,
                              void* d_out, int out_size, void* d_ws, size_t ws_size,
                              hipStream_t stream) {
    const float* in_embed  = (const float*)d_in[0];   // [VOCAB, 128] f32
    const float* out_embed = (const float*)d_in[1];   // [VOCAB, 128] f32
    const int*   context   = (const int*)d_in[2];     // [B, 8]
    const int*   target    = (const int*)d_in[3];     // [B]
    const int*   negatives = (const int*)d_in[4];     // [B, 10]
    float*       out       = (float*)d_out;           // [B]

    const int B = in_sizes[3];                        // 16384
    const int grid = (B + GRP - 1) / GRP;             // 1024 blocks, 1 wave each

    cbow_neg_wmma_kernel<<<grid, 32, 0, stream>>>(in_embed, out_embed,
                                                  context, target, negatives, out);
}